// PromptAnchorBank_22591527977505
// MI455X (gfx1250) — compile-verified
//
#include <hip/hip_runtime.h>
#include <hip/hip_bf16.h>
#include <math.h>

// ---------------------------------------------------------------------------
// PromptAnchorBank for MI455X (gfx1250, wave32)
//
// Stage A: partial token sums (1 GiB streamed once, NT-hint b128 loads)
// Stage B: reduce partials + L2-normalize desc -> desc_norm [64,1024]
// Stage C: L2-normalize anchors into zero-padded [16,1024] (rows 8..15 = 0)
// Stage D: sim = desc_norm @ anchors_norm^T via V_WMMA_F32_16X16X4_F32,
//          per-row argmax(sim) == argmin(1-sim)  -> ids [64]
// Stage E: sequential 64-step EMA scan in one workgroup (LDS-resident anchors)
// ---------------------------------------------------------------------------

typedef float v2f __attribute__((ext_vector_type(2)));
typedef float v4f __attribute__((ext_vector_type(4)));
typedef float v8f __attribute__((ext_vector_type(8)));

#define B_DIM   64
#define N_TOK   4096
#define C_DIM   1024
#define K_ANCH  8
#define K_PAD   16                 // anchors zero-padded to 16 rows for WMMA
#define NSPLIT  16                 // N split into 16 slices of 256 tokens
#define TOK_PER_SLICE (N_TOK / NSPLIT)

// -------------------- Stage A: partial sums over token slices --------------
// Pure single-pass stream: non-temporal loads so the 1 GiB input does not
// thrash the 192 MB L2 (nothing re-reads it).
__global__ void pab_partial_sum(const float* __restrict__ tok,
                                float* __restrict__ partials) {
    const int b  = blockIdx.x;          // 0..63
    const int s  = blockIdx.y;          // 0..15
    const int c4 = threadIdx.x * 4;     // 256 threads * 4 floats = 1024 chans
    const float* base = tok + ((size_t)b * N_TOK + (size_t)s * TOK_PER_SLICE) * C_DIM + c4;
    v4f acc = {};
#pragma unroll 4
    for (int n = 0; n < TOK_PER_SLICE; ++n) {
        const v4f v = __builtin_nontemporal_load(
            reinterpret_cast<const v4f*>(base + (size_t)n * C_DIM));
        acc += v;
    }
    *reinterpret_cast<v4f*>(partials + ((size_t)(b * NSPLIT + s)) * C_DIM + c4) = acc;
}

// ---------- Stage B: final reduce, scale by 1/N, L2-normalize rows ---------
__global__ void pab_reduce_norm_desc(const float* __restrict__ partials,
                                     float* __restrict__ desc_norm) {
    __shared__ float red[256];
    const int b   = blockIdx.x;
    const int tid = threadIdx.x;
    const int c4  = tid * 4;
    v4f s = {};
#pragma unroll
    for (int i = 0; i < NSPLIT; ++i) {
        s += *reinterpret_cast<const v4f*>(
            partials + ((size_t)(b * NSPLIT + i)) * C_DIM + c4);
    }
    s *= (1.0f / (float)N_TOK);
    red[tid] = s.x * s.x + s.y * s.y + s.z * s.z + s.w * s.w;
    __syncthreads();
    for (int off = 128; off > 0; off >>= 1) {
        if (tid < off) red[tid] += red[tid + off];
        __syncthreads();
    }
    const float inv = 1.0f / fmaxf(sqrtf(red[0]), 1e-12f);
    *reinterpret_cast<v4f*>(desc_norm + (size_t)b * C_DIM + c4) = s * inv;
}

// ------- Stage C: L2-normalize anchors into zero-padded 16-row matrix ------
__global__ void pab_norm_anchors(const float* __restrict__ anchors,
                                 float* __restrict__ anchors_norm) {
    __shared__ float red[256];
    const int k   = blockIdx.x;          // 0..15
    const int tid = threadIdx.x;
    const int c4  = tid * 4;
    if (k >= K_ANCH) {                   // zero pad rows 8..15 (block-uniform)
        v4f z = {};
        *reinterpret_cast<v4f*>(anchors_norm + (size_t)k * C_DIM + c4) = z;
        return;
    }
    const v4f v = *reinterpret_cast<const v4f*>(anchors + (size_t)k * C_DIM + c4);
    red[tid] = v.x * v.x + v.y * v.y + v.z * v.z + v.w * v.w;
    __syncthreads();
    for (int off = 128; off > 0; off >>= 1) {
        if (tid < off) red[tid] += red[tid + off];
        __syncthreads();
    }
    const float inv = 1.0f / fmaxf(sqrtf(red[0]), 1e-12f);
    *reinterpret_cast<v4f*>(anchors_norm + (size_t)k * C_DIM + c4) = v * inv;
}

// ---------- Stage D: WMMA cosine-sim GEMM (fp32) + per-row argmax ----------
// One wave per 16-row M tile.  A = desc_norm tile [16 x K],
// B = anchors_norm^T [K x 16] (rows of anchors_norm; cols 8..15 are zero by
// construction).  K stepped by 4 with V_WMMA_F32_16X16X4_F32.
//
// ISA 32-bit operand maps (CDNA5 ISA §7.12.2):
//   A 16x4 : lanes 0-15 -> M=lane, K={k0,k0+1}; lanes 16-31 -> K={k0+2,k0+3}
//   B 4x16 : lanes 0-15 -> N=lane, K rows {k0,k0+1}; lanes 16-31 -> {k0+2,k0+3}
//   D      : VGPR r, lanes 0-15 -> (M=r, N=lane); lanes 16-31 -> (M=8+r, N=lane-16)
__global__ void pab_sim_argmin(const float* __restrict__ desc_norm,
                               const float* __restrict__ anchors_norm,
                               int* __restrict__ ids_int,
                               float* __restrict__ ids_f) {
    __shared__ float sim[16 * 16];
    const int lane = threadIdx.x;        // 0..31
    const int tile = blockIdx.x;         // 0..3
    const int half = lane >> 4;          // 0 = lanes 0-15, 1 = lanes 16-31
    const int l16  = lane & 15;
    const int koff = half * 2;
    const float* aRow = desc_norm    + (size_t)(tile * 16 + l16) * C_DIM + koff;
    const float* bRow = anchors_norm + (size_t)l16 * C_DIM + koff;  // padded row

    v8f acc = {};                        // zero C accumulator
    for (int k0 = 0; k0 < C_DIM; k0 += 4) {
        // even-float offsets -> 8B aligned -> single global_load_b64 each
        const v2f a  = *reinterpret_cast<const v2f*>(aRow + k0);
        const v2f bb = *reinterpret_cast<const v2f*>(bRow + k0);
        // (neg_a, A, neg_b, B, c_mod, C, reuse_a, reuse_b)
        acc = __builtin_amdgcn_wmma_f32_16x16x4_f32(
            false, a, false, bb, (short)0, acc, false, false);
    }
#pragma unroll
    for (int r = 0; r < 8; ++r) {
        const int mm = half * 8 + r;
        sim[mm * 16 + l16] = acc[r];
    }
    __syncthreads();
    if (lane < 16) {
        const int row = tile * 16 + lane;
        float best = -3.0e38f;
        int bk = 0;
        for (int k = 0; k < K_ANCH; ++k) {
            const float s = sim[lane * 16 + k];
            if (s > best) { best = s; bk = k; }   // first-max == argmin(1-sim)
        }
        ids_int[row] = bk;
        ids_f[row]   = (float)bk;
    }
}

// ------------- Stage E: sequential per-sample EMA scan (1 WG) --------------
__global__ void __launch_bounds__(1024)
pab_ema_scan(const float* __restrict__ desc_norm,
             const float* __restrict__ anchors0,
             const float* __restrict__ counts_in,
             const int*   __restrict__ ids,
             float* __restrict__ out_anchors,
             float* __restrict__ out_counts) {
    __shared__ float a_l[K_ANCH * C_DIM];   // 32 KB of the 320 KB WGP LDS
    __shared__ float red[1024];
    __shared__ float cnt_l[K_ANCH];
    const int tid = threadIdx.x;            // 0..1023, one thread per channel
#pragma unroll
    for (int k = 0; k < K_ANCH; ++k)
        a_l[k * C_DIM + tid] = anchors0[(size_t)k * C_DIM + tid];
    if (tid < K_ANCH) cnt_l[tid] = counts_in[tid];
    __syncthreads();

    for (int b = 0; b < B_DIM; ++b) {
        const int k   = ids[b];                       // uniform across WG
        const float c = cnt_l[k];                     // stable since last barrier
        const float mom = (c < 1.0f) ? 0.0f : 0.9f;
        const float v = mom * a_l[k * C_DIM + tid]
                      + (1.0f - mom) * desc_norm[(size_t)b * C_DIM + tid];
        red[tid] = v * v;
        __syncthreads();
        for (int off = 512; off > 0; off >>= 1) {     // fixed-order -> deterministic
            if (tid < off) red[tid] += red[tid + off];
            __syncthreads();
        }
        const float inv = 1.0f / fmaxf(sqrtf(red[0]), 1e-12f);
        a_l[k * C_DIM + tid] = v * inv;
        __syncthreads();                              // red[0] read before reuse
        if (tid == 0) cnt_l[k] += 1.0f;
        __syncthreads();
    }

#pragma unroll
    for (int k = 0; k < K_ANCH; ++k)
        out_anchors[(size_t)k * C_DIM + tid] = a_l[k * C_DIM + tid];
    if (tid == 0)
        for (int k = 0; k < K_ANCH; ++k) out_counts[k] = cnt_l[k];
}

// ---------------------------------------------------------------------------
extern "C" void kernel_launch(void* const* d_in, const int* in_sizes, int n_in,
                              void* d_out, int out_size, void* d_ws, size_t ws_size,
                              hipStream_t stream) {
    (void)in_sizes; (void)n_in; (void)out_size; (void)ws_size;

    const float* tok     = (const float*)d_in[0];   // [64,4096,1024] f32
    const float* anchors = (const float*)d_in[1];   // [8,1024] f32
    const float* counts  = (const float*)d_in[2];   // [8] f32

    // workspace layout (floats)
    float* ws           = (float*)d_ws;
    float* partials     = ws;                                        // 64*16*1024
    float* desc_norm    = partials + (size_t)B_DIM * NSPLIT * C_DIM; // 64*1024
    float* anchors_norm = desc_norm + (size_t)B_DIM * C_DIM;         // 16*1024 (padded)
    int*   ids_int      = (int*)(anchors_norm + (size_t)K_PAD * C_DIM); // 64

    // output layout: [ids(64) | anchors(8192) | counts(8)] as float
    float* out          = (float*)d_out;
    float* out_ids      = out;
    float* out_anchors  = out + B_DIM;
    float* out_counts   = out + B_DIM + K_ANCH * C_DIM;

    pab_partial_sum<<<dim3(B_DIM, NSPLIT), 256, 0, stream>>>(tok, partials);
    pab_reduce_norm_desc<<<B_DIM, 256, 0, stream>>>(partials, desc_norm);
    pab_norm_anchors<<<K_PAD, 256, 0, stream>>>(anchors, anchors_norm);
    pab_sim_argmin<<<4, 32, 0, stream>>>(desc_norm, anchors_norm, ids_int, out_ids);
    pab_ema_scan<<<1, 1024, 0, stream>>>(desc_norm, anchors, counts, ids_int,
                                         out_anchors, out_counts);
}